// SpikeFP32Exp_76312978916076
// MI455X (gfx1250) — compile-verified
//
#include <hip/hip_runtime.h>

// Soft-logic "float bits -> 5-bit int" decode. Pure streaming kernel:
// 512MB in / 80MB out @ 23.3 TB/s => ~25us floor. Compute is negligible.
// Data movement uses the CDNA5 async LDS<->global path in both directions:
//   global_load_async_to_lds_b128  (coalesced 64B-per-group staging)
//   global_store_async_from_lds_b32 (perfectly coalesced output)
// synchronized with s_wait_asynccnt (ASYNCcnt).

typedef __attribute__((ext_vector_type(4))) float v4f;

#define GPB 256  // groups per block (== threads per block)

__device__ __forceinline__ unsigned int bit23(float f) {
    // inputs are exactly 0.0f or 1.0f; bit 23 of the FP32 encoding of 1.0f is 1
    return (__float_as_uint(f) >> 23) & 1u;
}

template <bool FULL>
__global__ __launch_bounds__(256) void spike_decode_kernel(
    const float* __restrict__ x, float* __restrict__ out, int ngroups)
{
    // Input stage: 80B per group (64B data + 16B pad -> conflict-free
    // ds_load_b128 at stride 80B: lane offsets L*20 cover all 64 banks).
    __shared__ __align__(16) unsigned char tile[GPB * 80];
    // Output stage: 5 dwords per group; stride 5 is coprime with 64 banks.
    __shared__ __align__(16) float otile[GPB * 5];

    const int t  = (int)threadIdx.x;
    const int g0 = (int)blockIdx.x * GPB;

    int nloc = GPB;
    if (!FULL) { nloc = ngroups - g0; if (nloc > GPB) nloc = GPB; }
    const int nchunks = nloc * 4;   // 16B chunks to stage

    const unsigned int lds_in =
        (unsigned int)(unsigned long long)(const void*)&tile[0];
    const unsigned int lds_out =
        (unsigned int)(unsigned long long)(const void*)&otile[0];
    const unsigned long long gbase =
        (unsigned long long)(const void*)x + (unsigned long long)g0 * 128ull;

    // ---- async stage global -> LDS: consecutive lanes -> consecutive chunks
    //      (each wave instruction covers 8 groups x 64B contiguous segments)
#pragma unroll
    for (int k = 0; k < 4; ++k) {
        const int j = t + k * 256;
        if (FULL || j < nchunks) {
            const int gl = j >> 2;       // local group
            const int c  = j & 3;        // 16B chunk within group (first 64B)
            unsigned long long gaddr =
                gbase + (unsigned long long)(gl * 128 + c * 16);
            unsigned int laddr = lds_in + (unsigned int)(gl * 80 + c * 16);
            asm volatile("global_load_async_to_lds_b128 %0, %1, off"
                         :: "v"(laddr), "v"(gaddr) : "memory");
        }
    }
    asm volatile("s_wait_asynccnt 0" ::: "memory");
    __syncthreads();

    // ---- compute from LDS: one group per thread --------------------------
    if (FULL || g0 + t < ngroups) {
        const v4f* lp = (const v4f*)&tile[t * 80];
        const v4f c0 = lp[0];   // elements 0..3
        const v4f c1 = lp[1];   // elements 4..7
        const v4f c2 = lp[2];   // elements 8..11
        const v4f c3 = lp[3];   // elements 12..15

        // E: exponent bits x[1..8], MSB first
        const unsigned int E =
            (bit23(c0.y) << 7) | (bit23(c0.z) << 6) | (bit23(c0.w) << 5) |
            (bit23(c1.x) << 4) | (bit23(c1.y) << 3) | (bit23(c1.z) << 2) |
            (bit23(c1.w) << 1) |  bit23(c2.x);
        const unsigned int s = (E + 129u) & 255u;   // E - 127 (mod 256)
        // M = 1.m0 m1 m2 m3 with m = x[9..12]
        const unsigned int M = 16u | (bit23(c2.y) << 3) | (bit23(c2.z) << 2) |
                                     (bit23(c2.w) << 1) |  bit23(c3.x);
        const unsigned int r = (s <= 4u) ? (((M << s) >> 4) & 31u) : 0u;

        float* o = &otile[t * 5];
        o[0] = (float)((r >> 4) & 1u);
        o[1] = (float)((r >> 3) & 1u);
        o[2] = (float)((r >> 2) & 1u);
        o[3] = (float)((r >> 1) & 1u);
        o[4] = (float)( r       & 1u);
    }
    __syncthreads();

    // ---- async store LDS -> global: 1280 dwords per block, consecutive
    //      lanes write consecutive dwords (128B per wave instruction).
    //      No explicit drain: S_ENDPGM performs an implicit S_WAIT_IDLE.
    const unsigned long long obase =
        (unsigned long long)(void*)out + (unsigned long long)g0 * 20ull;
#pragma unroll
    for (int k = 0; k < 5; ++k) {
        const int j = t + k * 256;
        if (FULL || j < nloc * 5) {
            unsigned long long gaddr = obase + (unsigned long long)(j * 4);
            unsigned int laddr = lds_out + (unsigned int)(j * 4);
            asm volatile("global_store_async_from_lds_b32 %0, %1, off"
                         :: "v"(gaddr), "v"(laddr) : "memory");
        }
    }
}

extern "C" void kernel_launch(void* const* d_in, const int* in_sizes, int n_in,
                              void* d_out, int out_size, void* d_ws, size_t ws_size,
                              hipStream_t stream) {
    const float* x = (const float*)d_in[0];
    float* out     = (float*)d_out;
    (void)n_in; (void)out_size; (void)d_ws; (void)ws_size;

    const int ngroups = in_sizes[0] / 32;          // 1024*4096 = 4,194,304
    const int blocks  = (ngroups + GPB - 1) / GPB; // 16384
    if (ngroups % GPB == 0) {
        spike_decode_kernel<true ><<<blocks, 256, 0, stream>>>(x, out, ngroups);
    } else {
        spike_decode_kernel<false><<<blocks, 256, 0, stream>>>(x, out, ngroups);
    }
}